// ChamferDistance_74955769249969
// MI455X (gfx1250) — compile-verified
//
#include <hip/hip_runtime.h>
#include <math.h>

typedef __attribute__((ext_vector_type(2))) float v2f;
typedef __attribute__((ext_vector_type(8))) float v8f;

#define BLOCK_THREADS 256
#define WAVES_PER_BLOCK 8
#define ROWS_PER_BLOCK (WAVES_PER_BLOCK * 16)
#define MAX_N 4096

// For each point x_i (16 rows per wave), compute min_j ||x_i - y_j||^2 using
// V_WMMA_F32_16X16X4_F32:
//   A = [x0,x1,x2,1]       (fixed per wave)
//   B = [-2y0,-2y1,-2y2,yd] (one ds_load_b64 per tile, pre-swizzled in LDS)
//   C = xd_m broadcast      (loop-invariant, built once via bpermute)
// => D = xd + yd - 2 x.y = full squared-distance tile, 1 WMMA per tile.
__global__ __launch_bounds__(BLOCK_THREADS) void chamfer_min_kernel(
    const float* __restrict__ X, const float* __restrict__ Y,
    float* __restrict__ outmin, int N)
{
    // Pre-swizzled B fragments: [tile][lane] -> float2 this lane feeds to WMMA.
    // lanes 0-15: (K0,K1) = (-2y0,-2y1); lanes 16-31: (K2,K3) = (-2y2, yd)
    __shared__ v2f blds[2 * MAX_N];  // 64 KB

    const int b   = blockIdx.y;
    const int tid = threadIdx.x;

    const float* Yb = Y + (size_t)b * N * 3;
    for (int p = tid; p < N; p += BLOCK_THREADS) {
        float y0 = Yb[p * 3 + 0];
        float y1 = Yb[p * 3 + 1];
        float y2 = Yb[p * 3 + 2];
        float yd = y0 * y0 + y1 * y1 + y2 * y2;
        int tile = p >> 4, n = p & 15;
        v2f lo; lo.x = -2.0f * y0; lo.y = -2.0f * y1;
        v2f hi; hi.x = -2.0f * y2; hi.y = yd;
        blds[tile * 32 + n]      = lo;
        blds[tile * 32 + 16 + n] = hi;
    }
    __syncthreads();

    const int lane = tid & 31;
    const int m    = lane & 15;   // row index within 16-tile
    const int half = lane >> 4;
    const int i0   = blockIdx.x * ROWS_PER_BLOCK + (tid >> 5) * 16;

    // A tile (16x4 f32): lanes 0-15 = (x0,x1), lanes 16-31 = (x2, 1)
    const float* xp = X + ((size_t)b * N + (i0 + m)) * 3;
    float x0 = xp[0], x1 = xp[1], x2 = xp[2];
    float xd = x0 * x0 + x1 * x1 + x2 * x2;
    v2f amat;
    amat.x = half ? x2 : x0;
    amat.y = half ? 1.0f : x1;

    // Loop-invariant C: C[v] at this lane = xd of row (v + 8*half).
    // xd of row r lives in lanes r and r+16, so a lane-indexed shuffle fetches it.
    v8f c;
#pragma unroll
    for (int v = 0; v < 8; ++v) c[v] = __shfl(xd, v + 8 * half, 32);

    v8f rm;
#pragma unroll
    for (int v = 0; v < 8; ++v) rm[v] = 3.0e38f;

    const int ntiles = N >> 4;
    for (int jt = 0; jt < ntiles; ++jt) {
        v2f bmat = blds[jt * 32 + lane];
        v8f d = __builtin_amdgcn_wmma_f32_16x16x4_f32(
            /*neg_a=*/false, amat, /*neg_b=*/false, bmat,
            /*c_mod=*/(short)0, c, /*reuse_a=*/false, /*reuse_b=*/false);
#pragma unroll
        for (int v = 0; v < 8; ++v) rm[v] = fminf(rm[v], d[v]);
    }

    // Cross-lane min within each 16-lane half (masks < 16 never cross halves).
#pragma unroll
    for (int mask = 1; mask <= 8; mask <<= 1) {
#pragma unroll
        for (int v = 0; v < 8; ++v)
            rm[v] = fminf(rm[v], __shfl_xor(rm[v], mask, 32));
    }

    // Lane 0 holds mins for rows i0+0..7, lane 16 for rows i0+8..15.
    if (m == 0) {
        float* dst = outmin + (size_t)b * N + i0 + half * 8;
#pragma unroll
        for (int v = 0; v < 8; ++v) dst[v] = rm[v];
    }
}

__global__ __launch_bounds__(BLOCK_THREADS) void chamfer_reduce_kernel(
    const float* __restrict__ mins, int count, float* __restrict__ out, float inv)
{
    __shared__ float sdata[BLOCK_THREADS];
    float s = 0.0f;
    for (int i = threadIdx.x; i < count; i += BLOCK_THREADS) s += mins[i];
    sdata[threadIdx.x] = s;
    __syncthreads();
    for (int off = BLOCK_THREADS / 2; off > 0; off >>= 1) {
        if (threadIdx.x < off) sdata[threadIdx.x] += sdata[threadIdx.x + off];
        __syncthreads();
    }
    if (threadIdx.x == 0) out[0] = sdata[0] * inv;
}

extern "C" void kernel_launch(void* const* d_in, const int* in_sizes, int n_in,
                              void* d_out, int out_size, void* d_ws, size_t ws_size,
                              hipStream_t stream)
{
    (void)n_in; (void)out_size; (void)ws_size;
    const float* p1 = (const float*)d_in[0];
    const float* p2 = (const float*)d_in[1];

    const int B = 8, D = 3;
    const int N = in_sizes[0] / (B * D);  // 4096

    float* mins = (float*)d_ws;  // 2*B*N floats = 256 KB scratch

    dim3 grid(N / ROWS_PER_BLOCK, B);  // (32, 8)
    // min over axis=2: for each x_i, nearest y_j
    chamfer_min_kernel<<<grid, BLOCK_THREADS, 0, stream>>>(p1, p2, mins, N);
    // min over axis=1: for each y_j, nearest x_i  (row-min of P^T)
    chamfer_min_kernel<<<grid, BLOCK_THREADS, 0, stream>>>(p2, p1,
                                                           mins + (size_t)B * N, N);
    // mean(axis=1 mins) + mean(axis=2 mins) = sum(all) / (B*N)
    chamfer_reduce_kernel<<<1, BLOCK_THREADS, 0, stream>>>(
        mins, 2 * B * N, (float*)d_out, 1.0f / (float)(B * N));
}